// ChannelExchangeWithConv_58634893525085
// MI455X (gfx1250) — compile-verified
//
#include <hip/hip_runtime.h>

typedef __attribute__((ext_vector_type(2))) float v2f;
typedef __attribute__((ext_vector_type(8))) float v8f;

#define HWPX   (512 * 512)     // pixels per channel
#define CFULL  128             // full channel count
#define CHC    64              // channels seen by each conv (C/p, p=2)
#define KSTEPS 16              // 64 / 4 (K per WMMA = 4)
#define TPW    8               // pixel tiles (of 16) per wave

// ---------------------------------------------------------------------------
// Kernel 1: pass-through copy of the odd channels of both tensors (float4).
// out layout: [out_lst (128*HW) | out_gui (128*HW)]
// ---------------------------------------------------------------------------
__global__ __launch_bounds__(256)
void ChannelExchange_copy_kernel(const float* __restrict__ lst,
                                 const float* __restrict__ gui,
                                 float* __restrict__ out) {
    const size_t per = (size_t)CHC * HWPX / 4;   // float4 count of odd chans per tensor
    size_t i = (size_t)blockIdx.x * blockDim.x + threadIdx.x;
    if (i >= 2 * per) return;

    const float4* src;
    float4*       dst;
    size_t idx = i;
    if (idx < per) {
        src = (const float4*)lst;
        dst = (float4*)out;
    } else {
        idx -= per;
        src = (const float4*)gui;
        dst = (float4*)(out + (size_t)CFULL * HWPX);
    }
    const size_t q4  = HWPX / 4;                 // float4 per channel
    size_t ch  = idx / q4;                       // 0..63 -> odd channel 2*ch+1
    size_t off = idx - ch * q4;
    size_t pos = (2 * ch + 1) * q4 + off;
    dst[pos] = src[pos];
}

// ---------------------------------------------------------------------------
// Kernel 2: 1x1-conv channel exchange via V_WMMA_F32_16X16X4_F32.
//   z==0: out_lst[::2] = w2 * gui[::2] + b2
//   z==1: out_gui[::2] = w1 * lst[::2] + b1
// One wave computes a 16(out-ch) x 16(px) tile, looping over TPW pixel tiles
// with the 16x64 weight tile held in VGPRs.
// ---------------------------------------------------------------------------
__global__ __launch_bounds__(32)
void ChannelExchangeWithConv_wmma_kernel(const float* __restrict__ lst,
                                         const float* __restrict__ gui,
                                         const float* __restrict__ w1,
                                         const float* __restrict__ b1,
                                         const float* __restrict__ w2,
                                         const float* __restrict__ b2,
                                         float* __restrict__ out) {
    const int lane = threadIdx.x;        // 0..31
    const int half = lane >> 4;          // 0 | 1  (selects K pair)
    const int l16  = lane & 15;

    const float* X;  const float* W;  const float* Bv;  float* O;
    if (blockIdx.z == 0) { X = gui; W = w2; Bv = b2; O = out; }
    else                 { X = lst; W = w1; Bv = b1; O = out + (size_t)CFULL * HWPX; }

    const int m0 = blockIdx.y * 16;      // out-channel tile base (conv space 0..63)

    // --- A tile: W[m0 + l16][0..63] as 16 v2f (K pairs; halves split K 0,1 vs 2,3)
    v2f a[KSTEPS];
    {
        const float* wrow = W + (size_t)(m0 + l16) * CHC;
        #pragma unroll
        for (int kk = 0; kk < KSTEPS; ++kk) {
            const int col = 4 * kk + 2 * half;
            a[kk] = *(const v2f*)(wrow + col);       // contiguous float2
        }
    }

    // --- bias per accumulator row: row(r) = m0 + r + 8*half
    float biasv[8];
    #pragma unroll
    for (int r = 0; r < 8; ++r) biasv[r] = Bv[m0 + r + 8 * half];

    const int p0base = blockIdx.x * (16 * TPW);

    for (int t = 0; t < TPW; ++t) {
        const int p0 = p0base + 16 * t;

        // --- B tile: X[::2][4kk..4kk+3][p0..p0+15]; lane = pixel, halves = K pair
        v2f bm[KSTEPS];
        #pragma unroll
        for (int kk = 0; kk < KSTEPS; ++kk) {
            const int k0 = 4 * kk + 2 * half;                    // conv-space in-channel
            const float* xp = X + (size_t)(2 * k0) * HWPX + p0 + l16;
            bm[kk].x = xp[0];
            bm[kk].y = xp[2 * HWPX];                             // next even channel
        }

        // --- accumulate: C initialized with bias, 16 chained f32 WMMAs (K=64)
        v8f c;
        #pragma unroll
        for (int r = 0; r < 8; ++r) c[r] = biasv[r];
        #pragma unroll
        for (int kk = 0; kk < KSTEPS; ++kk) {
            c = __builtin_amdgcn_wmma_f32_16x16x4_f32(
                    /*neg_a=*/false, a[kk], /*neg_b=*/false, bm[kk],
                    /*c_mod=*/(short)0, c, /*reuse_a=*/false, /*reuse_b=*/false);
        }

        // --- store: D VGPR r -> out channel 2*(m0 + r + 8*half), pixel p0+l16
        #pragma unroll
        for (int r = 0; r < 8; ++r) {
            const int row = m0 + r + 8 * half;
            O[(size_t)(2 * row) * HWPX + p0 + l16] = c[r];
        }
    }
}

extern "C" void kernel_launch(void* const* d_in, const int* in_sizes, int n_in,
                              void* d_out, int out_size, void* d_ws, size_t ws_size,
                              hipStream_t stream) {
    const float* lst = (const float*)d_in[0];
    const float* gui = (const float*)d_in[1];
    const float* w1  = (const float*)d_in[2];
    const float* b1  = (const float*)d_in[3];
    const float* w2  = (const float*)d_in[4];
    const float* b2  = (const float*)d_in[5];
    // d_in[6] == p (always 2 per setup_inputs)
    float* out = (float*)d_out;

    // pass-through copy of odd channels: 2 * 64 * HW / 4 float4's
    {
        const size_t total = 2ull * CHC * HWPX / 4;
        const int threads = 256;
        const int blocks = (int)((total + threads - 1) / threads);   // 32768
        ChannelExchange_copy_kernel<<<blocks, threads, 0, stream>>>(lst, gui, out);
    }

    // WMMA conv-exchange: grid (pixel-tile groups, M tiles, conv id)
    {
        dim3 grid(HWPX / (16 * TPW), CHC / 16, 2);                   // (2048, 4, 2)
        ChannelExchangeWithConv_wmma_kernel<<<grid, 32, 0, stream>>>(
            lst, gui, w1, b1, w2, b2, out);
    }
}